// GATConv_62225486185198
// MI455X (gfx1250) — compile-verified
//
#include <hip/hip_runtime.h>
#include <math.h>

typedef __attribute__((ext_vector_type(2))) float v2f;
typedef __attribute__((ext_vector_type(8))) float v8f;

#define FEATS 128      // F = H*D = 128
#define HEADS 4
#define DPH   32
#define ROWS  64       // nodes per block in GEMM kernel
#define LDF   132      // padded LDS row stride (floats): conflict-free b64 frags
#define NEG_SLOPE 0.2f

// ---------------------------------------------------------------------------
// Kernel 1: fused projection GEMM (h = feat @ W^T) via V_WMMA_F32_16X16X4_F32
//           + el/er attention logits computed from the LDS h-tile.
// Block: 256 threads = 8 waves. Block tile: 64 nodes x 128 cols.
// Wave w owns cols [16w, 16w+16) across 4 row-tiles of 16 nodes.
// ---------------------------------------------------------------------------
__global__ __launch_bounds__(256)
void gat_gemm_kernel(const float* __restrict__ feat,
                     const float* __restrict__ W,       // [128, 128] row-major (o, k)
                     const float* __restrict__ attn_l,  // [H*D] = 128
                     const float* __restrict__ attn_r,  // [H*D] = 128
                     float* __restrict__ h,             // [N, 128]
                     float* __restrict__ el,            // [N, H]
                     float* __restrict__ er,            // [N, H]
                     int n_nodes)
{
    __shared__ __align__(16) float tile[ROWS * LDF];

    const int t    = threadIdx.x;
    const int wave = t >> 5;
    const int lane = t & 31;
    const int base = blockIdx.x * ROWS;

    // Cooperative, coalesced load of the feat tile [ROWS x 128] (float4 units).
    for (int i = t; i < ROWS * (FEATS / 4); i += 256) {
        int row = i >> 5;            // 32 float4 per row
        int c4  = (i & 31) << 2;
        float4 v = make_float4(0.f, 0.f, 0.f, 0.f);
        if (base + row < n_nodes)
            v = *(const float4*)(feat + (size_t)(base + row) * FEATS + c4);
        *(float4*)(&tile[row * LDF + c4]) = v;
    }
    __syncthreads();

    const int colb = wave * 16;          // this wave's 16 output columns
    const int lm   = lane & 15;          // M (A) / N (B) index for this lane
    const int kh   = (lane >> 4) << 1;   // K sub-offset: lanes 0-15 -> 0, 16-31 -> 2

    v8f z = {};
    v8f acc[4];
    acc[0] = z; acc[1] = z; acc[2] = z; acc[3] = z;

    // Lane's B column: row (colb+lm) of W, i.e. column of W^T. W is 64KB -> L2-hot.
    const float* wrow = W + (size_t)(colb + lm) * FEATS;

    for (int kb = 0; kb < FEATS; kb += 4) {
        const int k0 = kb + kh;
        // B fragment: B[k0..k0+1][colb+lm] = W[colb+lm][k0..k0+1]  (8B aligned)
        v2f b = *(const v2f*)(wrow + k0);
        #pragma unroll
        for (int rt = 0; rt < 4; ++rt) {
            // A fragment: feat[base + rt*16 + lm][k0..k0+1] from LDS (ds_load_b64)
            v2f a = *(const v2f*)(&tile[(rt * 16 + lm) * LDF + k0]);
            acc[rt] = __builtin_amdgcn_wmma_f32_16x16x4_f32(
                false, a, false, b, (short)0, acc[rt], false, false);
        }
    }

    __syncthreads();   // everyone done reading the feat tile; reuse it for h

    // Scatter C tiles: VGPR r holds M = r (+8 for upper half-wave), N = lane&15.
    #pragma unroll
    for (int rt = 0; rt < 4; ++rt) {
        #pragma unroll
        for (int r = 0; r < 8; ++r) {
            int row = rt * 16 + r + ((lane >> 4) << 3);
            float v = acc[rt][r];
            tile[row * LDF + colb + lm] = v;
            if (base + row < n_nodes)
                h[(size_t)(base + row) * FEATS + colb + lm] = v;
        }
    }
    __syncthreads();

    // Fused attention logits: 256 threads -> 64 nodes x 4 heads.
    {
        int row  = t >> 2;
        int head = t & 3;
        if (base + row < n_nodes) {
            float sl = 0.f, sr = 0.f;
            #pragma unroll 8
            for (int d = 0; d < DPH; ++d) {
                float hv = tile[row * LDF + head * DPH + d];
                sl += hv * attn_l[head * DPH + d];
                sr += hv * attn_r[head * DPH + d];
            }
            el[(size_t)(base + row) * HEADS + head] = sl;
            er[(size_t)(base + row) * HEADS + head] = sr;
        }
    }
}

// ---------------------------------------------------------------------------
// Kernel 2: init out = 0, segment-max = -inf, segment-sum = 0
// ---------------------------------------------------------------------------
__global__ void gat_init_kernel(float* __restrict__ out, float* __restrict__ mx,
                                float* __restrict__ sm, int out_n, int nh)
{
    int i = blockIdx.x * blockDim.x + threadIdx.x;
    if (i < out_n) out[i] = 0.f;
    if (i < nh) { mx[i] = -INFINITY; sm[i] = 0.f; }
}

__device__ __forceinline__ float leaky(float v) {
    return v > 0.f ? v : NEG_SLOPE * v;
}

// Order-independent float atomic max via int-max / uint-min trick.
__device__ __forceinline__ void atomicMaxF(float* addr, float v) {
    if (v >= 0.f) atomicMax((int*)addr, __float_as_int(v));
    else          atomicMin((unsigned int*)addr, __float_as_uint(v));
}

// ---------------------------------------------------------------------------
// Kernel 3: per-(edge, head) score -> segment max over dst
// ---------------------------------------------------------------------------
__global__ void gat_edge_max_kernel(const int* __restrict__ src, const int* __restrict__ dst,
                                    const float* __restrict__ el, const float* __restrict__ er,
                                    float* __restrict__ mx, int eh)
{
    int i = blockIdx.x * blockDim.x + threadIdx.x;
    if (i >= eh) return;
    int e  = i >> 2;
    int hd = i & 3;
    int sn = src[e], dn = dst[e];
    float v = leaky(el[sn * HEADS + hd] + er[dn * HEADS + hd]);
    atomicMaxF(&mx[dn * HEADS + hd], v);
}

// ---------------------------------------------------------------------------
// Kernel 4: per-(edge, head) exp(e - m[dst]) -> segment sum over dst
// ---------------------------------------------------------------------------
__global__ void gat_edge_sum_kernel(const int* __restrict__ src, const int* __restrict__ dst,
                                    const float* __restrict__ el, const float* __restrict__ er,
                                    const float* __restrict__ mx, float* __restrict__ sm, int eh)
{
    int i = blockIdx.x * blockDim.x + threadIdx.x;
    if (i >= eh) return;
    int e  = i >> 2;
    int hd = i & 3;
    int sn = src[e], dn = dst[e];
    float v = leaky(el[sn * HEADS + hd] + er[dn * HEADS + hd]);
    atomicAdd(&sm[dn * HEADS + hd], expf(v - mx[dn * HEADS + hd]));
}

// ---------------------------------------------------------------------------
// Kernel 5: weighted aggregation: out[dst] += alpha * h[src]
// One thread per (edge, channel); a wave covers one edge (uniform src/dst).
// ---------------------------------------------------------------------------
__global__ void gat_edge_aggregate_kernel(const int* __restrict__ src, const int* __restrict__ dst,
                                          const float* __restrict__ el, const float* __restrict__ er,
                                          const float* __restrict__ mx, const float* __restrict__ sm,
                                          const float* __restrict__ h, float* __restrict__ out,
                                          int total)
{
    int i = blockIdx.x * blockDim.x + threadIdx.x;
    if (i >= total) return;
    int e  = i >> 7;          // edge
    int j  = i & 127;         // channel = head*32 + d
    int hd = j >> 5;
    int sn = src[e], dn = dst[e];
    float v     = leaky(el[sn * HEADS + hd] + er[dn * HEADS + hd]);
    float alpha = expf(v - mx[dn * HEADS + hd]) / sm[dn * HEADS + hd];
    atomicAdd(&out[(size_t)dn * FEATS + j], alpha * h[(size_t)sn * FEATS + j]);
}

// ---------------------------------------------------------------------------
extern "C" void kernel_launch(void* const* d_in, const int* in_sizes, int n_in,
                              void* d_out, int out_size, void* d_ws, size_t ws_size,
                              hipStream_t stream)
{
    const float* feat   = (const float*)d_in[0];   // [N, 128]
    const float* W      = (const float*)d_in[1];   // [128, 128]
    const float* attn_l = (const float*)d_in[2];   // [1, 4, 32]
    const float* attn_r = (const float*)d_in[3];   // [1, 4, 32]
    const int*   src    = (const int*)d_in[4];     // [E]
    const int*   dst    = (const int*)d_in[5];     // [E]
    float*       out    = (float*)d_out;           // [N, 4, 32]

    const int N = in_sizes[0] / FEATS;
    const int E = in_sizes[4];
    const int NH = N * HEADS;

    // Workspace layout
    float* h_ws  = (float*)d_ws;                   // N*128
    float* el_ws = h_ws  + (size_t)N * FEATS;      // N*4
    float* er_ws = el_ws + (size_t)NH;             // N*4
    float* mx_ws = er_ws + (size_t)NH;             // N*4
    float* sm_ws = mx_ws + (size_t)NH;             // N*4

    // 1) init out / max / sum
    {
        int mx_n = out_size > NH ? out_size : NH;
        gat_init_kernel<<<(mx_n + 255) / 256, 256, 0, stream>>>(out, mx_ws, sm_ws, out_size, NH);
    }
    // 2) projection GEMM + el/er (WMMA f32)
    {
        int blocks = (N + ROWS - 1) / ROWS;
        gat_gemm_kernel<<<blocks, 256, 0, stream>>>(feat, W, attn_l, attn_r,
                                                    h_ws, el_ws, er_ws, N);
    }
    // 3) segment max
    {
        int eh = E * HEADS;
        gat_edge_max_kernel<<<(eh + 255) / 256, 256, 0, stream>>>(src, dst, el_ws, er_ws, mx_ws, eh);
    }
    // 4) segment sum of exp
    {
        int eh = E * HEADS;
        gat_edge_sum_kernel<<<(eh + 255) / 256, 256, 0, stream>>>(src, dst, el_ws, er_ws,
                                                                  mx_ws, sm_ws, eh);
    }
    // 5) weighted aggregation
    {
        int total = E * FEATS;   // 204.8M threads
        gat_edge_aggregate_kernel<<<(total + 255) / 256, 256, 0, stream>>>(
            src, dst, el_ws, er_ws, mx_ws, sm_ws, h_ws, out, total);
    }
}